// BottleneckBoxConv_43336220016854
// MI455X (gfx1250) — compile-verified
//
#include <hip/hip_runtime.h>
#include <hip/hip_bf16.h>

#define B_    4
#define C_    128
#define CB_   32
#define H_    160
#define W_    160
#define HW_   (H_ * W_)
#define KOUT_ 128
#define EPS_  1e-5f

typedef __attribute__((ext_vector_type(2))) float v2f;
typedef __attribute__((ext_vector_type(8))) float v8f;

// ---------------------------------------------------------------------------
// 1) GEMM: hpre[b,o,hw] = sum_c w1[o,c] * x[b,c,hw]  via V_WMMA_F32_16X16X4_F32
//    One wave -> 32x16 output tile (two 16x16 accumulators), K=128 in steps of 4.
//    f32 A 16x4 layout: lanes 0-15 hold (K=k0,k0+1), lanes 16-31 hold (K=k0+2,k0+3)
//    f32 C/D layout:   VGPR r, lanes 0-15 = row r, lanes 16-31 = row r+8
// ---------------------------------------------------------------------------
__global__ void __launch_bounds__(256) k_gemm(const float* __restrict__ x,
                                              const float* __restrict__ w1,
                                              float* __restrict__ hpre) {
  const int wave = threadIdx.x >> 5;
  const int lane = threadIdx.x & 31;
  const int half = lane >> 4;          // which K/row half of the wave
  const int l    = lane & 15;
  const int tile = blockIdx.x * 8 + wave;          // 6400 tiles total
  const int b    = tile / (HW_ / 16);
  const long n0  = (long)(tile % (HW_ / 16)) * 16;

  v8f acc0 = {};
  v8f acc1 = {};
  const float* xb = x + (size_t)b * C_ * HW_ + n0 + l;

#pragma unroll 4
  for (int k0 = 0; k0 < C_; k0 += 4) {
    const int ka = k0 + half * 2;
    v2f a0, a1, bb;
    a0.x = w1[l * C_ + ka];
    a0.y = w1[l * C_ + ka + 1];
    a1.x = w1[(16 + l) * C_ + ka];
    a1.y = w1[(16 + l) * C_ + ka + 1];
    const float* xp = xb + (size_t)ka * HW_;
    bb.x = xp[0];
    bb.y = xp[HW_];
    acc0 = __builtin_amdgcn_wmma_f32_16x16x4_f32(false, a0, false, bb,
                                                 (short)0, acc0, false, false);
    acc1 = __builtin_amdgcn_wmma_f32_16x16x4_f32(false, a1, false, bb,
                                                 (short)0, acc1, false, false);
  }

  float* hp = hpre + (size_t)b * CB_ * HW_ + n0 + l;
#pragma unroll
  for (int r = 0; r < 8; ++r) {
    const int m = r + half * 8;
    hp[(size_t)m * HW_]        = acc0[r];
    hp[(size_t)(16 + m) * HW_] = acc1[r];
  }
}

// ---------------------------------------------------------------------------
// 2/6) Per-channel sum & sum-of-squares (train-mode BN stats), deterministic:
//      one block per channel, LDS tree reduction, no atomics.
// ---------------------------------------------------------------------------
__global__ void __launch_bounds__(256) k_stats(const float* __restrict__ data,
                                               float* __restrict__ stats, int C) {
  __shared__ float s1[256];
  __shared__ float s2[256];
  const int c = blockIdx.x;
  float sum = 0.f, sq = 0.f;
  for (int b = 0; b < B_; ++b) {
    const float* p = data + ((size_t)b * C + c) * HW_;
    for (int i = threadIdx.x; i < HW_; i += 256) {
      const float v = p[i];
      sum += v;
      sq += v * v;
    }
  }
  s1[threadIdx.x] = sum;
  s2[threadIdx.x] = sq;
  __syncthreads();
  for (int off = 128; off > 0; off >>= 1) {
    if ((int)threadIdx.x < off) {
      s1[threadIdx.x] += s1[threadIdx.x + off];
      s2[threadIdx.x] += s2[threadIdx.x + off];
    }
    __syncthreads();
  }
  if (threadIdx.x == 0) {
    stats[c]     = s1[0];
    stats[C + c] = s2[0];
  }
}

// ---------------------------------------------------------------------------
// 3) BN1 + ReLU + horizontal prefix sum into padded integral image
//    II shape (B, CB, H+1, W+1); row i+1, col 0 = 0.
// ---------------------------------------------------------------------------
__global__ void __launch_bounds__(256) k_prefixW(const float* __restrict__ hpre,
                                                 const float* __restrict__ stats1,
                                                 const float* __restrict__ g1,
                                                 const float* __restrict__ b1,
                                                 float* __restrict__ II) {
  const int t = blockIdx.x * 256 + threadIdx.x;   // B*CB*H rows
  if (t >= B_ * CB_ * H_) return;
  const int i = t % H_;
  const int c = (t / H_) % CB_;
  const int b = t / (H_ * CB_);
  const float n    = (float)(B_ * HW_);
  const float mean = stats1[c] / n;
  const float var  = stats1[CB_ + c] / n - mean * mean;
  const float inv  = rsqrtf(var + EPS_);
  const float ga = g1[c], be = b1[c];
  const float* hp = hpre + (((size_t)b * CB_ + c) * H_ + i) * W_;
  float* ip = II + (((size_t)b * CB_ + c) * (H_ + 1) + (i + 1)) * (W_ + 1);
  ip[0] = 0.f;
  float run = 0.f;
  for (int j = 0; j < W_; ++j) {
    float v = ga * (hp[j] - mean) * inv + be;
    v = v > 0.f ? v : 0.f;
    run += v;
    ip[j + 1] = run;
  }
}

// ---------------------------------------------------------------------------
// 4) Vertical prefix over II (threads adjacent in column index -> coalesced).
// ---------------------------------------------------------------------------
__global__ void __launch_bounds__(256) k_prefixH(float* __restrict__ II) {
  const int t = blockIdx.x * 256 + threadIdx.x;   // B*CB*(W+1) columns
  if (t >= B_ * CB_ * (W_ + 1)) return;
  const int j = t % (W_ + 1);
  const int c = (t / (W_ + 1)) % CB_;
  const int b = t / ((W_ + 1) * CB_);
  float* base = II + (((size_t)b * CB_ + c) * (H_ + 1)) * (W_ + 1) + j;
  base[0] = 0.f;
  float run = 0.f;
  for (int i = 1; i <= H_; ++i) {
    run += base[(size_t)i * (W_ + 1)];
    base[(size_t)i * (W_ + 1)] = run;
  }
}

// ---------------------------------------------------------------------------
// 5) Box conv: 4-corner inclusion/exclusion with bilinear fractional taps.
// ---------------------------------------------------------------------------
__device__ __forceinline__ float bilin(const float* __restrict__ IIc,
                                       int u0, int u1, float au,
                                       int v0, int v1, float av) {
  const float* r0p = IIc + (size_t)u0 * (W_ + 1);
  const float* r1p = IIc + (size_t)u1 * (W_ + 1);
  const float r0 = r0p[v0] * (1.f - av) + r0p[v1] * av;
  const float r1 = r1p[v0] * (1.f - av) + r1p[v1] * av;
  return r0 * (1.f - au) + r1 * au;
}

__global__ void __launch_bounds__(W_) k_box(const float* __restrict__ II,
                                            const float* __restrict__ xmin,
                                            const float* __restrict__ xmax,
                                            const float* __restrict__ ymin,
                                            const float* __restrict__ ymax,
                                            float* __restrict__ boxout) {
  const int i = blockIdx.x % H_;
  const int k = (blockIdx.x / H_) % KOUT_;
  const int b = blockIdx.x / (H_ * KOUT_);
  const int j = threadIdx.x;
  if (j >= W_) return;
  const int cc = k >> 2;   // nb == 4 boxes per channel

  const float ulo_off = xmin[k];
  const float uhi_off = xmax[k] + 1.f;
  const float vlo_off = ymin[k];
  const float vhi_off = ymax[k] + 1.f;

  auto mk = [](float coord, float lim, int& p0, int& p1, float& a) {
    coord = fminf(fmaxf(coord, 0.f), lim);
    const float f = floorf(coord);
    a  = coord - f;
    p0 = (int)f;
    p1 = p0 + 1;
    if (p1 > (int)lim) p1 = (int)lim;
  };

  int ul0, ul1, uh0, uh1, vl0, vl1, vh0, vh1;
  float aul, auh, avl, avh;
  mk((float)i + ulo_off, (float)H_, ul0, ul1, aul);
  mk((float)i + uhi_off, (float)H_, uh0, uh1, auh);
  mk((float)j + vlo_off, (float)W_, vl0, vl1, avl);
  mk((float)j + vhi_off, (float)W_, vh0, vh1, avh);

  const float* IIc = II + (((size_t)b * CB_ + cc) * (H_ + 1)) * (W_ + 1);
  const float val = bilin(IIc, uh0, uh1, auh, vh0, vh1, avh)
                  - bilin(IIc, ul0, ul1, aul, vh0, vh1, avh)
                  - bilin(IIc, uh0, uh1, auh, vl0, vl1, avl)
                  + bilin(IIc, ul0, ul1, aul, vl0, vl1, avl);
  boxout[(((size_t)b * KOUT_ + k) * H_ + i) * W_ + j] = val;
}

// ---------------------------------------------------------------------------
// 7) BN2 + residual + ReLU
// ---------------------------------------------------------------------------
__global__ void __launch_bounds__(256) k_final(const float* __restrict__ x,
                                               const float* __restrict__ boxout,
                                               const float* __restrict__ stats2,
                                               const float* __restrict__ g2,
                                               const float* __restrict__ b2,
                                               float* __restrict__ out) {
  const size_t t = (size_t)blockIdx.x * 256 + threadIdx.x;
  if (t >= (size_t)B_ * KOUT_ * HW_) return;
  const int c = (int)((t / HW_) % KOUT_);
  const float n    = (float)(B_ * HW_);
  const float mean = stats2[c] / n;
  const float var  = stats2[KOUT_ + c] / n - mean * mean;
  const float inv  = rsqrtf(var + EPS_);
  float v = x[t] + g2[c] * (boxout[t] - mean) * inv + b2[c];
  out[t] = v > 0.f ? v : 0.f;
}

extern "C" void kernel_launch(void* const* d_in, const int* in_sizes, int n_in,
                              void* d_out, int out_size, void* d_ws, size_t ws_size,
                              hipStream_t stream) {
  (void)in_sizes; (void)n_in; (void)out_size; (void)ws_size;
  const float* x    = (const float*)d_in[0];
  const float* w1   = (const float*)d_in[1];
  const float* g1   = (const float*)d_in[2];
  const float* b1   = (const float*)d_in[3];
  const float* xmin = (const float*)d_in[4];
  const float* xmax = (const float*)d_in[5];
  const float* ymin = (const float*)d_in[6];
  const float* ymax = (const float*)d_in[7];
  const float* g2   = (const float*)d_in[8];
  const float* b2   = (const float*)d_in[9];
  float* out = (float*)d_out;

  // Workspace layout (floats): hpre | II | boxout | stats1 | stats2  (~79 MB)
  float* ws     = (float*)d_ws;
  float* hpre   = ws;                                              // B*CB*HW
  float* II     = hpre + (size_t)B_ * CB_ * HW_;                   // B*CB*(H+1)*(W+1)
  float* boxout = II + (size_t)B_ * CB_ * (H_ + 1) * (W_ + 1);     // B*K*HW
  float* stats1 = boxout + (size_t)B_ * KOUT_ * HW_;               // 2*CB
  float* stats2 = stats1 + 2 * CB_;                                // 2*K

  k_gemm<<<(B_ * HW_ / 16) / 8, 256, 0, stream>>>(x, w1, hpre);
  k_stats<<<CB_, 256, 0, stream>>>(hpre, stats1, CB_);
  k_prefixW<<<(B_ * CB_ * H_ + 255) / 256, 256, 0, stream>>>(hpre, stats1, g1, b1, II);
  k_prefixH<<<(B_ * CB_ * (W_ + 1) + 255) / 256, 256, 0, stream>>>(II);
  k_box<<<B_ * KOUT_ * H_, W_, 0, stream>>>(II, xmin, xmax, ymin, ymax, boxout);
  k_stats<<<KOUT_, 256, 0, stream>>>(boxout, stats2, KOUT_);
  k_final<<<(int)(((size_t)B_ * KOUT_ * HW_ + 255) / 256), 256, 0, stream>>>(
      x, boxout, stats2, g2, b2, out);
}